// CustomDCNv4_2293512536475
// MI455X (gfx1250) — compile-verified
//
#include <hip/hip_runtime.h>

#define NB   16
#define CH   256
#define GR   8
#define CGR  32
#define HH   56
#define WW   56
#define HWSZ (HH*WW)
#define KGR  9
#define DOM  216
#define OMLD 224   // padded leading dim for om buffer (14 tiles of 16)

typedef __attribute__((ext_vector_type(2))) float v2f;
typedef __attribute__((ext_vector_type(8))) float v8f;

// ---------------------------------------------------------------------------
// GEMM with K-major A:  Out[n,p,co] = sum_k A[n][k][p] * Wm[co][k] + bias[co]
// A element (p,k) at A + n*K*HWSZ + k*HWSZ + p.
// Each wave computes a 32x32 tile = 2x2 WMMA sub-tiles (4 x v8f accumulators);
// per K-step: 2 A-frags + 2 B-frags -> 4 WMMAs (2 loads per WMMA per lane).
// blockDim = 64 (2 waves along M). grid = (HWSZ/64, Ncols/32, NB).
// ---------------------------------------------------------------------------
__global__ void k_value_gemm(const float* __restrict__ A,
                             const float* __restrict__ Wm,
                             const float* __restrict__ bias,
                             float* __restrict__ Out) {
    const int wave = threadIdx.x >> 5;
    const int lane = threadIdx.x & 31;
    const int half = lane >> 4;
    const int m    = lane & 15;
    const int pBase = (blockIdx.x * 2 + wave) * 32;
    const int nBase = blockIdx.y * 32;
    const int n     = blockIdx.z;
    const float* Ab = A + (size_t)n * CH * HWSZ;

    v8f acc00 = {}, acc01 = {}, acc10 = {}, acc11 = {};
#pragma unroll 4
    for (int k0 = 0; k0 < CH; k0 += 4) {
        const float* a0 = Ab + (size_t)(k0 + 2 * half) * HWSZ + pBase + m;
        v2f aF0, aF1, bF0, bF1;
        aF0.x = a0[0];        aF0.y = a0[HWSZ];
        aF1.x = a0[16];       aF1.y = a0[HWSZ + 16];
        const float* b0 = Wm + (size_t)(nBase + m) * CH + k0 + half;
        bF0.x = b0[0];        bF0.y = b0[2];
        const float* b1 = b0 + 16 * CH;
        bF1.x = b1[0];        bF1.y = b1[2];
        acc00 = __builtin_amdgcn_wmma_f32_16x16x4_f32(false, aF0, false, bF0,
                                                      (short)0, acc00, false, false);
        acc01 = __builtin_amdgcn_wmma_f32_16x16x4_f32(false, aF0, false, bF1,
                                                      (short)0, acc01, false, false);
        acc10 = __builtin_amdgcn_wmma_f32_16x16x4_f32(false, aF1, false, bF0,
                                                      (short)0, acc10, false, false);
        acc11 = __builtin_amdgcn_wmma_f32_16x16x4_f32(false, aF1, false, bF1,
                                                      (short)0, acc11, false, false);
    }
    const float bb0 = bias[nBase + m];
    const float bb1 = bias[nBase + 16 + m];
    const size_t row0 = (size_t)n * HWSZ + pBase + half * 8;
#pragma unroll
    for (int r = 0; r < 8; ++r) {
        Out[(row0 + r)      * CH + nBase + m]      = acc00[r] + bb0;
        Out[(row0 + r)      * CH + nBase + 16 + m] = acc01[r] + bb1;
        Out[(row0 + 16 + r) * CH + nBase + m]      = acc10[r] + bb0;
        Out[(row0 + 16 + r) * CH + nBase + 16 + m] = acc11[r] + bb1;
    }
}

// ---------------------------------------------------------------------------
// Depthwise 3x3 conv on cat(inp, ref), groups=C (2 in-ch per group).
// Writes K-major layout: dwb[n][c][p] so the om GEMM can reuse the K-major
// loader with no transpose. grid = (HH, CH, NB), blockDim = 64 (x dim).
// ---------------------------------------------------------------------------
__global__ void k_dwconv(const float* __restrict__ inp,
                         const float* __restrict__ ref,
                         const float* __restrict__ dw_w,
                         const float* __restrict__ dw_b,
                         float* __restrict__ dwb) {
    const int x = threadIdx.x;
    if (x >= WW) return;
    const int y = blockIdx.x;
    const int c = blockIdx.y;
    const int n = blockIdx.z;

    const int ch0 = 2 * c, ch1 = 2 * c + 1;
    const float* s0 = (ch0 < CH) ? inp + ((size_t)n * CH + ch0) * HWSZ
                                 : ref + ((size_t)n * CH + ch0 - CH) * HWSZ;
    const float* s1 = (ch1 < CH) ? inp + ((size_t)n * CH + ch1) * HWSZ
                                 : ref + ((size_t)n * CH + ch1 - CH) * HWSZ;
    float acc = dw_b[c];
#pragma unroll
    for (int ky = 0; ky < 3; ++ky) {
        const int yy = y + ky - 1;
        if (yy < 0 || yy >= HH) continue;
#pragma unroll
        for (int kx = 0; kx < 3; ++kx) {
            const int xx = x + kx - 1;
            if (xx < 0 || xx >= WW) continue;
            const float w0 = dw_w[((c * 2 + 0) * 3 + ky) * 3 + kx];
            const float w1 = dw_w[((c * 2 + 1) * 3 + ky) * 3 + kx];
            acc += w0 * s0[yy * WW + xx] + w1 * s1[yy * WW + xx];
        }
    }
    dwb[((size_t)n * CH + c) * HWSZ + y * WW + x] = acc;
}

// ---------------------------------------------------------------------------
// om GEMM: om[n,p,d] = sum_k dwb[n][k][p] * om_w[d][k] + om_b[d], d < 216,
// stored with padded ld = 224. 32x32 tile per wave.
// grid = (HWSZ/64, 224/32 = 7, NB), blockDim = 64.
// ---------------------------------------------------------------------------
__global__ void k_om_gemm(const float* __restrict__ A,
                          const float* __restrict__ Wm,
                          const float* __restrict__ bias,
                          float* __restrict__ Out) {
    const int wave = threadIdx.x >> 5;
    const int lane = threadIdx.x & 31;
    const int half = lane >> 4;
    const int m    = lane & 15;
    const int pBase = (blockIdx.x * 2 + wave) * 32;
    const int nBase = blockIdx.y * 32;
    const int n     = blockIdx.z;
    const int d0    = nBase + m;
    const int d1    = nBase + 16 + m;
    const bool v0   = (d0 < DOM);
    const bool v1   = (d1 < DOM);
    const float* Ab = A + (size_t)n * CH * HWSZ;

    v8f acc00 = {}, acc01 = {}, acc10 = {}, acc11 = {};
#pragma unroll 4
    for (int k0 = 0; k0 < CH; k0 += 4) {
        const float* a0 = Ab + (size_t)(k0 + 2 * half) * HWSZ + pBase + m;
        v2f aF0, aF1, bF0, bF1;
        aF0.x = a0[0];        aF0.y = a0[HWSZ];
        aF1.x = a0[16];       aF1.y = a0[HWSZ + 16];
        const float* b0 = Wm + (size_t)d0 * CH + k0 + half;
        const float* b1 = Wm + (size_t)d1 * CH + k0 + half;
        bF0.x = v0 ? b0[0] : 0.0f;   bF0.y = v0 ? b0[2] : 0.0f;
        bF1.x = v1 ? b1[0] : 0.0f;   bF1.y = v1 ? b1[2] : 0.0f;
        acc00 = __builtin_amdgcn_wmma_f32_16x16x4_f32(false, aF0, false, bF0,
                                                      (short)0, acc00, false, false);
        acc01 = __builtin_amdgcn_wmma_f32_16x16x4_f32(false, aF0, false, bF1,
                                                      (short)0, acc01, false, false);
        acc10 = __builtin_amdgcn_wmma_f32_16x16x4_f32(false, aF1, false, bF0,
                                                      (short)0, acc10, false, false);
        acc11 = __builtin_amdgcn_wmma_f32_16x16x4_f32(false, aF1, false, bF1,
                                                      (short)0, acc11, false, false);
    }
    const float bb0 = v0 ? bias[d0] : 0.0f;
    const float bb1 = v1 ? bias[d1] : 0.0f;
    const size_t row0 = (size_t)n * HWSZ + pBase + half * 8;
#pragma unroll
    for (int r = 0; r < 8; ++r) {
        Out[(row0 + r)      * OMLD + d0] = acc00[r] + bb0;
        Out[(row0 + r)      * OMLD + d1] = acc01[r] + bb1;
        Out[(row0 + 16 + r) * OMLD + d0] = acc10[r] + bb0;
        Out[(row0 + 16 + r) * OMLD + d1] = acc11[r] + bb1;
    }
}

// ---------------------------------------------------------------------------
// Deformable bilinear sampling. One wave32 per (n, p, g); lane = channel in
// group -> every gather is a contiguous 128B line. grid = N*HW*G/8, blk=256.
// ---------------------------------------------------------------------------
__global__ void k_sample(const float* __restrict__ xv,
                         const float* __restrict__ om,
                         float* __restrict__ samp) {
    const int wid  = (blockIdx.x * blockDim.x + threadIdx.x) >> 5;
    const int lane = threadIdx.x & 31;
    const int g = wid & (GR - 1);
    const int p = (wid >> 3) % HWSZ;
    const int n = wid / (GR * HWSZ);
    const int y = p / WW, x = p % WW;

    const float* omp = om + ((size_t)n * HWSZ + p) * OMLD + g * (KGR * 3);
    const float* xvb = xv + (size_t)n * HWSZ * CH + g * CGR + lane;

    float acc = 0.0f;
#pragma unroll
    for (int pk = 0; pk < KGR; ++pk) {
        const float offx = omp[pk * 2];
        const float offy = omp[pk * 2 + 1];
        const float mval = omp[2 * KGR + pk];
        const float iy = (float)(y + pk / 3 - 1) + offy;
        const float ix = (float)(x + pk % 3 - 1) + offx;
        const float y0 = floorf(iy), x0 = floorf(ix);
        const float fy = iy - y0,    fx = ix - x0;
        const int y0i = (int)y0, x0i = (int)x0;
#pragma unroll
        for (int dy = 0; dy < 2; ++dy) {
#pragma unroll
            for (int dx = 0; dx < 2; ++dx) {
                const int yy = y0i + dy, xx = x0i + dx;
                const float wgt = (dy ? fy : 1.0f - fy) * (dx ? fx : 1.0f - fx) * mval;
                if (yy >= 0 && yy < HH && xx >= 0 && xx < WW)
                    acc += wgt * xvb[(size_t)(yy * WW + xx) * CH];
            }
        }
    }
    samp[((size_t)n * HWSZ + p) * CH + g * CGR + lane] = acc;
}

// ---------------------------------------------------------------------------
// Output GEMM, row-major A: y[n,co,p] = sum_k samp[n,p,k]*out_w[co,k]+b[co],
// stored NCHW (transposed). 32x32 tile per wave.
// grid = (HWSZ/64, CH/32 = 8, NB), blockDim = 64.
// ---------------------------------------------------------------------------
__global__ void k_out_gemm(const float* __restrict__ A,
                           const float* __restrict__ Wm,
                           const float* __restrict__ bias,
                           float* __restrict__ Out) {
    const int wave = threadIdx.x >> 5;
    const int lane = threadIdx.x & 31;
    const int half = lane >> 4;
    const int m    = lane & 15;
    const int pBase = (blockIdx.x * 2 + wave) * 32;
    const int nBase = blockIdx.y * 32;
    const int n     = blockIdx.z;

    v8f acc00 = {}, acc01 = {}, acc10 = {}, acc11 = {};
#pragma unroll 4
    for (int k0 = 0; k0 < CH; k0 += 4) {
        const float* a0 = A + ((size_t)n * HWSZ + pBase + m) * CH + k0 + 2 * half;
        v2f aF0, aF1, bF0, bF1;
        aF0.x = a0[0];            aF0.y = a0[1];
        aF1.x = a0[16 * CH];      aF1.y = a0[16 * CH + 1];
        const float* b0 = Wm + (size_t)(nBase + m) * CH + k0 + half;
        bF0.x = b0[0];            bF0.y = b0[2];
        const float* b1 = b0 + 16 * CH;
        bF1.x = b1[0];            bF1.y = b1[2];
        acc00 = __builtin_amdgcn_wmma_f32_16x16x4_f32(false, aF0, false, bF0,
                                                      (short)0, acc00, false, false);
        acc01 = __builtin_amdgcn_wmma_f32_16x16x4_f32(false, aF0, false, bF1,
                                                      (short)0, acc01, false, false);
        acc10 = __builtin_amdgcn_wmma_f32_16x16x4_f32(false, aF1, false, bF0,
                                                      (short)0, acc10, false, false);
        acc11 = __builtin_amdgcn_wmma_f32_16x16x4_f32(false, aF1, false, bF1,
                                                      (short)0, acc11, false, false);
    }
    const float bb0 = bias[nBase + m];
    const float bb1 = bias[nBase + 16 + m];
    // D layout: lane col = nBase(+16)+m, rows M = half*8 + r -> NCHW store
    const size_t o00 = (size_t)n * CH * HWSZ + (size_t)(nBase + m) * HWSZ
                     + pBase + half * 8;
    const size_t o01 = (size_t)n * CH * HWSZ + (size_t)(nBase + 16 + m) * HWSZ
                     + pBase + half * 8;
#pragma unroll
    for (int r = 0; r < 8; ++r) {
        Out[o00 + r]      = acc00[r] + bb0;
        Out[o01 + r]      = acc01[r] + bb1;
        Out[o00 + 16 + r] = acc10[r] + bb0;
        Out[o01 + 16 + r] = acc11[r] + bb1;
    }
}

// ---------------------------------------------------------------------------
extern "C" void kernel_launch(void* const* d_in, const int* in_sizes, int n_in,
                              void* d_out, int out_size, void* d_ws, size_t ws_size,
                              hipStream_t stream) {
    const float* ref     = (const float*)d_in[0];
    const float* inp     = (const float*)d_in[1];
    const float* value_w = (const float*)d_in[2];
    const float* value_b = (const float*)d_in[3];
    const float* dw_w    = (const float*)d_in[4];
    const float* dw_b    = (const float*)d_in[5];
    const float* om_w    = (const float*)d_in[6];
    const float* om_b    = (const float*)d_in[7];
    const float* out_w   = (const float*)d_in[8];
    const float* out_b   = (const float*)d_in[9];
    float* out = (float*)d_out;

    float* xv   = (float*)d_ws;                       // N*HW*C
    float* dwb  = xv  + (size_t)NB * HWSZ * CH;       // N*HW*C (K-major dw)
    float* om   = dwb + (size_t)NB * HWSZ * CH;       // N*HW*224
    float* samp = dwb;                                // reuse: dwb dead after om GEMM

    // 1) value projection (WMMA f32, 32x32 tile per wave)
    k_value_gemm<<<dim3(HWSZ / 64, CH / 32, NB), 64, 0, stream>>>(inp, value_w, value_b, xv);
    // 2) depthwise 3x3 conv -> K-major
    k_dwconv<<<dim3(HH, CH, NB), 64, 0, stream>>>(inp, ref, dw_w, dw_b, dwb);
    // 3) offset/mask projection (WMMA f32, N padded 216->224)
    k_om_gemm<<<dim3(HWSZ / 64, OMLD / 32, NB), 64, 0, stream>>>(dwb, om_w, om_b, om);
    // 4) deformable bilinear sampling (1 wave per (n,p,g))
    k_sample<<<dim3((NB * HWSZ * GR) / 8), 256, 0, stream>>>(xv, om, samp);
    // 5) output projection (WMMA f32) + NCHW transpose store
    k_out_gemm<<<dim3(HWSZ / 64, CH / 32, NB), 64, 0, stream>>>(samp, out_w, out_b, out);
}